// DetectHead_4681514353330
// MI455X (gfx1250) — compile-verified
//
#include <hip/hip_runtime.h>
#include <hip/hip_bf16.h>

// DetectHead on MI455X: per-pixel GEMM (M=72ch, N=B*H*W pixels, K=1024) using
// V_WMMA_F32_16X16X4_F32 (fp32 in/out, matches the f32 reference; kernel is
// HBM-bound at ~34 FLOP/B so fp32 matrix path is the right precision).

typedef __attribute__((ext_vector_type(2))) float v2f;
typedef __attribute__((ext_vector_type(4))) float v4f;
typedef __attribute__((ext_vector_type(8))) float v8f;

#define BATCH 4
#define CIN   1024
#define HH    200
#define WWID  176
#define HWPIX (HH * WWID)      // 35200 pixels per image
#define OC    72               // 18 cls + 42 box + 12 dir (concat order)
#define OCP   80               // padded to 5 tiles of 16
#define MT    5                // M-tiles (80/16)
#define NT    2                // pixel tiles per wave (32 pixels)
#define PIX_PER_WAVE (NT * 16)
#define WAVES_PER_BLOCK 8

// ---- prep: pack [cls_w; box_w; dir_w] into zero-padded 80x1024 + bias[80] ----
__global__ void prep_pack(const float* __restrict__ cls_w,
                          const float* __restrict__ box_w,
                          const float* __restrict__ dir_w,
                          const float* __restrict__ cls_b,
                          const float* __restrict__ box_b,
                          const float* __restrict__ dir_b,
                          float* __restrict__ wpad,
                          float* __restrict__ bpad) {
  int idx = blockIdx.x * blockDim.x + threadIdx.x;
  if (idx < OCP * CIN) {
    int r = idx / CIN, c = idx % CIN;
    float v = 0.0f;
    if (r < 18)      v = cls_w[r * CIN + c];
    else if (r < 60) v = box_w[(r - 18) * CIN + c];
    else if (r < 72) v = dir_w[(r - 60) * CIN + c];
    wpad[idx] = v;
  }
  if (idx < OCP) {
    float v = 0.0f;
    if (idx < 18)      v = cls_b[idx];
    else if (idx < 60) v = box_b[idx - 18];
    else if (idx < 72) v = dir_b[idx - 60];
    bpad[idx] = v;
  }
}

// ---- main GEMM: each wave -> 80 channels x 32 pixels, K=1024 in steps of 4 ----
__global__ __launch_bounds__(256) void detect_head_wmma(
    const float* __restrict__ x,     // (B, 1024, H, W)
    const float* __restrict__ wpad,  // (80, 1024) row-major, zero padded
    const float* __restrict__ bpad,  // (80,)
    float* __restrict__ out) {       // (B, H, W, 72)
  const int lane = threadIdx.x & 31;
  const int wave = threadIdx.x >> 5;
  const int gw   = blockIdx.x * WAVES_PER_BLOCK + wave;

  const int waves_per_img = HWPIX / PIX_PER_WAVE;   // 1100, exact
  const int img = gw / waves_per_img;
  const int p0  = (gw % waves_per_img) * PIX_PER_WAVE;

  const int lo = lane & 15;   // M row (A) / N pixel (B,C,D)
  const int hi = lane >> 4;   // selects K pair (A,B) / M half (C,D)

  const float* __restrict__ xb = x + (size_t)img * CIN * HWPIX;

  v8f acc[MT][NT] = {};

  // Per-lane base pointers (loop-invariant parts)
  const float* __restrict__ apb = wpad + (size_t)lo * CIN + 2 * hi;   // + mt*16*CIN + k
  const float* __restrict__ bpb = xb + (size_t)(2 * hi) * HWPIX + p0 + lo; // + (k+v)*HW + nt*16

#pragma unroll 4
  for (int k = 0; k < CIN; k += 4) {
    // A fragments: 16x4 tile per M-tile; lane holds W[m][k+2*hi .. +1] (b64 load)
    v2f a[MT];
#pragma unroll
    for (int mt = 0; mt < MT; ++mt)
      a[mt] = *(const v2f*)(apb + (size_t)(mt * 16) * CIN + k);

    // B fragments: 4x16 tile per N-tile; VGPR v holds x[k + v + 2*hi][pixel]
    v2f bfr[NT];
#pragma unroll
    for (int nt = 0; nt < NT; ++nt) {
      const float* bp = bpb + (size_t)k * HWPIX + nt * 16;
      bfr[nt][0] = bp[0];       // K = k + 2*hi
      bfr[nt][1] = bp[HWPIX];   // K = k + 2*hi + 1
    }

#pragma unroll
    for (int mt = 0; mt < MT; ++mt)
#pragma unroll
      for (int nt = 0; nt < NT; ++nt)
        acc[mt][nt] = __builtin_amdgcn_wmma_f32_16x16x4_f32(
            /*neg_a=*/false, a[mt], /*neg_b=*/false, bfr[nt],
            /*c_mod=*/(short)0, acc[mt][nt],
            /*reuse_a=*/false, /*reuse_b=*/false);
  }

  // Epilogue: bias add + channels-last store.
  // Lane holds D[m = v + 8*hi][n = lo]; oc = mt*16 + 8*hi + v  -> 8 consecutive
  // channels per lane, 16B-aligned (pixel stride 288B, oc base multiple of 8).
#pragma unroll
  for (int mt = 0; mt < MT; ++mt) {
    const int ocb = mt * 16 + 8 * hi;
    if (ocb >= OC) continue;   // skip oc 72..79 pad half-tile (zeros anyway)
    const v4f bia0 = *(const v4f*)(bpad + ocb);
    const v4f bia1 = *(const v4f*)(bpad + ocb + 4);
#pragma unroll
    for (int nt = 0; nt < NT; ++nt) {
      const v8f r = acc[mt][nt];
      v4f r0 = { r[0] + bia0[0], r[1] + bia0[1], r[2] + bia0[2], r[3] + bia0[3] };
      v4f r1 = { r[4] + bia1[0], r[5] + bia1[1], r[6] + bia1[2], r[7] + bia1[3] };
      float* op = out + (size_t)(img * HWPIX + p0 + nt * 16 + lo) * OC + ocb;
      *(v4f*)op       = r0;
      *(v4f*)(op + 4) = r1;
    }
  }
}

extern "C" void kernel_launch(void* const* d_in, const int* in_sizes, int n_in,
                              void* d_out, int out_size, void* d_ws, size_t ws_size,
                              hipStream_t stream) {
  const float* x     = (const float*)d_in[0];
  const float* cls_w = (const float*)d_in[1];
  const float* cls_b = (const float*)d_in[2];
  const float* box_w = (const float*)d_in[3];
  const float* box_b = (const float*)d_in[4];
  const float* dir_w = (const float*)d_in[5];
  const float* dir_b = (const float*)d_in[6];
  float* out = (float*)d_out;

  float* wpad = (float*)d_ws;            // 80*1024 floats = 320 KB
  float* bpad = wpad + OCP * CIN;        // 80 floats

  prep_pack<<<(OCP * CIN + 255) / 256, 256, 0, stream>>>(
      cls_w, box_w, dir_w, cls_b, box_b, dir_b, wpad, bpad);

  const int total_waves = BATCH * (HWPIX / PIX_PER_WAVE);  // 4*1100 = 4400
  const int blocks = total_waves / WAVES_PER_BLOCK;        // 550, exact
  detect_head_wmma<<<blocks, 256, 0, stream>>>(x, wpad, bpad, out);
}